// DenseFlashAttention_21028159881270
// MI455X (gfx1250) — compile-verified
//
#include <hip/hip_runtime.h>
#include <math.h>

typedef __attribute__((ext_vector_type(2))) float v2f;
typedef __attribute__((ext_vector_type(8))) float v8f;

// ---------------------------------------------------------------------------
// Monotonic order-preserving float <-> uint key (finite floats): bigger float
// <=> bigger key. Lets us use hardware uint atomicMax for segment-max.
// ---------------------------------------------------------------------------
__device__ __forceinline__ unsigned fkey(float f) {
    unsigned u = __float_as_uint(f);
    return (u & 0x80000000u) ? ~u : (u | 0x80000000u);
}
__device__ __forceinline__ float fkey_inv(unsigned k) {
    unsigned u = (k & 0x80000000u) ? (k & 0x7FFFFFFFu) : ~k;
    return __uint_as_float(u);
}

// ---------------------------------------------------------------------------
// Zero / init workspace accumulators (harness poisons ws with 0xAA once).
// ---------------------------------------------------------------------------
__global__ void init_ws_kernel(unsigned* __restrict__ smaxkey,
                               float* __restrict__ Z,
                               float* __restrict__ outacc, int N) {
    int i = blockIdx.x * blockDim.x + threadIdx.x;
    if (i < N * 32) outacc[i] = 0.0f;
    if (i < N) { smaxkey[i] = 0u; Z[i] = 0.0f; }
}

// ---------------------------------------------------------------------------
// One 16x16x4 f32 WMMA k-step against two 16-col tiles of a 32x32 weight.
// B(4x16) fragment: VGPR j (j=0,1) of lane l holds W[kb + 2*(l/16) + j][n],
// n = l%16 (+16 for the second tile).
// ---------------------------------------------------------------------------
__device__ __forceinline__ void wmma_step(const float* __restrict__ W, int kb,
                                          int half, int m, v2f a,
                                          v8f& acc0, v8f& acc1) {
    int r0 = (kb + 2 * half + 0) * 32;
    int r1 = (kb + 2 * half + 1) * 32;
    v2f b0, b1;
    b0.x = W[r0 + m];        b0.y = W[r1 + m];
    b1.x = W[r0 + 16 + m];   b1.y = W[r1 + 16 + m];
    acc0 = __builtin_amdgcn_wmma_f32_16x16x4_f32(false, a, false, b0,
                                                 (short)0, acc0, false, false);
    acc1 = __builtin_amdgcn_wmma_f32_16x16x4_f32(false, a, false, b1,
                                                 (short)0, acc1, false, false);
}

// ---------------------------------------------------------------------------
// Q/K/V projection: one wave computes one 16-row tile of all three outputs.
// A(16x4 f32) fragment: lane l holds x[row0 + l%16][kb + 2*(l/16) + {0,1}].
// C/D(16x16 f32): VGPR r of lane l holds D[r + 8*(l/16)][l%16].
// ---------------------------------------------------------------------------
__global__ __launch_bounds__(256) void qkv_wmma_kernel(
    const float* __restrict__ x,
    const float* __restrict__ Wq, const float* __restrict__ Wk,
    const float* __restrict__ Wv,
    float* __restrict__ Q, float* __restrict__ K, float* __restrict__ V,
    int ntiles) {
    int wave = (int)((blockIdx.x * blockDim.x + threadIdx.x) >> 5);
    if (wave >= ntiles) return;               // wave-uniform: EXEC stays all-1s
    int lane = (int)(threadIdx.x & 31u);
    int half = lane >> 4;
    int m    = lane & 15;
    size_t row0 = (size_t)wave * 16;

    v8f q0 = {}, q1 = {}, k0 = {}, k1 = {}, v0 = {}, v1 = {};

#pragma unroll
    for (int kk = 0; kk < 8; ++kk) {
        int kb = kk * 4;
        const float* ap = x + (row0 + m) * 32 + kb + 2 * half;
        v2f a; a.x = ap[0]; a.y = ap[1];      // contiguous pair -> b64 load
        wmma_step(Wq, kb, half, m, a, q0, q1);
        wmma_step(Wk, kb, half, m, a, k0, k1);
        wmma_step(Wv, kb, half, m, a, v0, v1);
    }

#pragma unroll
    for (int r = 0; r < 8; ++r) {
        size_t row = (row0 + r + 8 * half) * 32;
        Q[row + m]      = q0[r];  Q[row + 16 + m] = q1[r];
        K[row + m]      = k0[r];  K[row + 16 + m] = k1[r];
        V[row + m]      = v0[r];  V[row + 16 + m] = v1[r];
    }
}

// Scalar tail for N % 16 rows (0 blocks for N = 100000).
__global__ void qkv_tail_kernel(const float* __restrict__ x,
                                const float* __restrict__ Wq,
                                const float* __restrict__ Wk,
                                const float* __restrict__ Wv,
                                float* __restrict__ Q, float* __restrict__ K,
                                float* __restrict__ V, int row_start, int N) {
    int idx = blockIdx.x * blockDim.x + threadIdx.x;
    int nrows = N - row_start;
    if (idx >= nrows * 32) return;
    int row = row_start + idx / 32, col = idx & 31;
    float q = 0.f, k = 0.f, v = 0.f;
    for (int t = 0; t < 32; ++t) {
        float xv = x[(size_t)row * 32 + t];
        q += xv * Wq[t * 32 + col];
        k += xv * Wk[t * 32 + col];
        v += xv * Wv[t * 32 + col];
    }
    size_t o = (size_t)row * 32 + col;
    Q[o] = q; K[o] = k; V[o] = v;
}

// ---------------------------------------------------------------------------
// Per-edge scores + segment max (uint-key atomicMax, hits L2 atomic units).
// ---------------------------------------------------------------------------
__global__ void edge_score_kernel(const float* __restrict__ Q,
                                  const float* __restrict__ K,
                                  const int* __restrict__ ei,
                                  float* __restrict__ scores,
                                  unsigned* __restrict__ smaxkey, int E) {
    int e = blockIdx.x * blockDim.x + threadIdx.x;
    if (e >= E) return;
    int s = ei[e], r = ei[E + e];
    const float4* q = (const float4*)(Q + (size_t)r * 32);
    const float4* k = (const float4*)(K + (size_t)s * 32);
    float acc = 0.f;
#pragma unroll
    for (int i = 0; i < 8; ++i) {
        float4 a = q[i], b = k[i];
        acc += a.x * b.x + a.y * b.y + a.z * b.z + a.w * b.w;
    }
    float sc = acc * 0.17677669529663687f;    // 1/sqrt(32)
    scores[e] = sc;
    atomicMax(&smaxkey[r], fkey(sc));
}

// ---------------------------------------------------------------------------
// alpha = exp(score - smax[recv]); Z[recv] += alpha;
// outacc[recv] += alpha * V[send]  (normalization deferred: commutes with @Wo)
// ---------------------------------------------------------------------------
__global__ void edge_agg_kernel(const float* __restrict__ V,
                                const int* __restrict__ ei,
                                const float* __restrict__ scores,
                                const unsigned* __restrict__ smaxkey,
                                float* __restrict__ Z,
                                float* __restrict__ outacc, int E) {
    int e = blockIdx.x * blockDim.x + threadIdx.x;
    if (e >= E) return;
    int s = ei[e], r = ei[E + e];
    float alpha = __expf(scores[e] - fkey_inv(smaxkey[r]));
    atomicAdd(&Z[r], alpha);
    const float4* v = (const float4*)(V + (size_t)s * 32);
    float* o = outacc + (size_t)r * 32;
#pragma unroll
    for (int i = 0; i < 8; ++i) {
        float4 vv = v[i];
        atomicAdd(o + 4 * i + 0, alpha * vv.x);
        atomicAdd(o + 4 * i + 1, alpha * vv.y);
        atomicAdd(o + 4 * i + 2, alpha * vv.z);
        atomicAdd(o + 4 * i + 3, alpha * vv.w);
    }
}

// ---------------------------------------------------------------------------
// out = x + ((outacc / (Z+1e-6)) @ Wo); row scale folded into A fragment,
// residual fused into store. One wave per 16-row tile.
// ---------------------------------------------------------------------------
__global__ __launch_bounds__(256) void out_wmma_kernel(
    const float* __restrict__ x, const float* __restrict__ outacc,
    const float* __restrict__ Z, const float* __restrict__ Wo,
    float* __restrict__ out, int ntiles) {
    int wave = (int)((blockIdx.x * blockDim.x + threadIdx.x) >> 5);
    if (wave >= ntiles) return;
    int lane = (int)(threadIdx.x & 31u);
    int half = lane >> 4;
    int m    = lane & 15;
    size_t row0 = (size_t)wave * 16;

    float invz = 1.0f / (Z[row0 + m] + 1e-6f);   // per A-row scalar (row = m)
    v8f c0 = {}, c1 = {};

#pragma unroll
    for (int kk = 0; kk < 8; ++kk) {
        int kb = kk * 4;
        const float* ap = outacc + (row0 + m) * 32 + kb + 2 * half;
        v2f a; a.x = ap[0] * invz; a.y = ap[1] * invz;
        wmma_step(Wo, kb, half, m, a, c0, c1);
    }

#pragma unroll
    for (int r = 0; r < 8; ++r) {
        size_t row = (row0 + r + 8 * half) * 32;
        out[row + m]      = x[row + m]      + c0[r];
        out[row + 16 + m] = x[row + 16 + m] + c1[r];
    }
}

__global__ void out_tail_kernel(const float* __restrict__ x,
                                const float* __restrict__ outacc,
                                const float* __restrict__ Z,
                                const float* __restrict__ Wo,
                                float* __restrict__ out, int row_start, int N) {
    int idx = blockIdx.x * blockDim.x + threadIdx.x;
    int nrows = N - row_start;
    if (idx >= nrows * 32) return;
    int row = row_start + idx / 32, col = idx & 31;
    float invz = 1.0f / (Z[row] + 1e-6f);
    float acc = 0.f;
    for (int t = 0; t < 32; ++t)
        acc += outacc[(size_t)row * 32 + t] * invz * Wo[t * 32 + col];
    size_t o = (size_t)row * 32 + col;
    out[o] = x[o] + acc;
}

// ---------------------------------------------------------------------------
extern "C" void kernel_launch(void* const* d_in, const int* in_sizes, int n_in,
                              void* d_out, int out_size, void* d_ws,
                              size_t ws_size, hipStream_t stream) {
    const float* x  = (const float*)d_in[0];
    const int*   ei = (const int*)d_in[1];          // int32 [2, E] (JAX x64 off)
    const float* Wq = (const float*)d_in[2];
    const float* Wk = (const float*)d_in[3];
    const float* Wv = (const float*)d_in[4];
    const float* Wo = (const float*)d_in[5];
    float* out = (float*)d_out;

    const int N = in_sizes[0] / 32;
    const int E = in_sizes[1] / 2;

    // Workspace layout (~58.5 MB): Q | K | V | outacc | scores | smaxkey | Z
    float* Q       = (float*)d_ws;
    float* K       = Q + (size_t)N * 32;
    float* V       = K + (size_t)N * 32;
    float* outacc  = V + (size_t)N * 32;
    float* scores  = outacc + (size_t)N * 32;
    unsigned* smaxkey = (unsigned*)(scores + E);
    float* Z       = (float*)(smaxkey + N);

    const int TB = 256;
    int ntiles = N / 16;
    int rem_row0 = ntiles * 16;
    int rem = N - rem_row0;

    // 0) init accumulators
    {
        int total = N * 32;
        init_ws_kernel<<<(total + TB - 1) / TB, TB, 0, stream>>>(smaxkey, Z,
                                                                 outacc, N);
    }
    // 1) Q/K/V projection (WMMA f32 16x16x4)
    {
        int threads = ntiles * 32;
        if (threads > 0)
            qkv_wmma_kernel<<<(threads + TB - 1) / TB, TB, 0, stream>>>(
                x, Wq, Wk, Wv, Q, K, V, ntiles);
        if (rem > 0)
            qkv_tail_kernel<<<(rem * 32 + TB - 1) / TB, TB, 0, stream>>>(
                x, Wq, Wk, Wv, Q, K, V, rem_row0, N);
    }
    // 2) per-edge scores + segment max
    edge_score_kernel<<<(E + TB - 1) / TB, TB, 0, stream>>>(Q, K, ei, scores,
                                                            smaxkey, E);
    // 3) per-edge exp + segment sum + weighted V aggregation
    edge_agg_kernel<<<(E + TB - 1) / TB, TB, 0, stream>>>(V, ei, scores,
                                                          smaxkey, Z, outacc, E);
    // 4) out = x + (outacc/Z) @ Wo (WMMA, scale + residual fused)
    {
        int threads = ntiles * 32;
        if (threads > 0)
            out_wmma_kernel<<<(threads + TB - 1) / TB, TB, 0, stream>>>(
                x, outacc, Z, Wo, out, ntiles);
        if (rem > 0)
            out_tail_kernel<<<(rem * 32 + TB - 1) / TB, TB, 0, stream>>>(
                x, outacc, Z, Wo, out, rem_row0, N);
    }
}